// GETD_45208825757879
// MI455X (gfx1250) — compile-verified
//
#include <hip/hip_runtime.h>
#include <hip/hip_bf16.h>

#define DE     10
#define RANK   40
#define PADM   48            // 40 padded to 3 x 16 WMMA tiles
#define STRIDE 52            // LDS row stride (floats): bank-conflict-free, 8B aligned
#define BUFSZ  (PADM * STRIDE)
#define NT     3             // 16x16 tiles per dim
#define KSTEPS 12            // 48 / 4
#define EPS_BN 1e-5f
#define NTHREADS 128         // 4 wave32s per workgroup

typedef __attribute__((ext_vector_type(2))) float v2f;
typedef __attribute__((ext_vector_type(8))) float v8f;

// dynamic LDS bytes: vsh(7*16) + 10 matrix buffers + red(128)
#define SMEM_FLOATS (7 * 16 + 10 * BUFSZ + NTHREADS)
#define SMEM_BYTES  (SMEM_FLOATS * 4)

// One 16x16 output tile of dst = A @ B (48x48 operands in LDS, stride 52),
// accumulated over K=48 with 12 chained V_WMMA_F32_16X16X4_F32.
__device__ inline void wmma_tile(float* __restrict__ dst,
                                 const float* __restrict__ A,
                                 const float* __restrict__ Bm,
                                 int tm, int tn, int lane) {
    const int r    = lane & 15;
    const int koff = (lane >> 4) << 1;   // lanes 0-15: K{0,1}; lanes 16-31: K{2,3}
    v8f acc = {};
#pragma unroll
    for (int kb = 0; kb < KSTEPS; ++kb) {
        const float* ap = A + (tm * 16 + r) * STRIDE + kb * 4 + koff;
        v2f a;
        a.x = ap[0];
        a.y = ap[1];
        const float* bp = Bm + (kb * 4 + koff) * STRIDE + tn * 16 + r;
        v2f bv;
        bv.x = bp[0];
        bv.y = bp[STRIDE];
        acc = __builtin_amdgcn_wmma_f32_16x16x4_f32(
                false, a, false, bv, (short)0, acc, false, false);
    }
    // C/D layout: VGPR g -> row g (lanes 0-15) / row g+8 (lanes 16-31)
    const int n       = tn * 16 + r;
    const int rowbase = tm * 16 + ((lane >> 4) << 3);
#pragma unroll
    for (int g = 0; g < 8; ++g)
        dst[(rowbase + g) * STRIDE + n] = acc[g];
}

extern "C" __global__ void __launch_bounds__(NTHREADS)
getd_tr_kernel(const int* __restrict__ r_idx,
               const int* __restrict__ e1_idx, const int* __restrict__ e2_idx,
               const int* __restrict__ e3_idx, const int* __restrict__ e4_idx,
               const int* __restrict__ e5_idx, const int* __restrict__ e6_idx,
               const float* __restrict__ E, const float* __restrict__ R,
               const float* __restrict__ Z0, const float* __restrict__ Z1,
               const float* __restrict__ Z2, const float* __restrict__ Z3,
               const float* __restrict__ Z4, const float* __restrict__ Z5,
               const float* __restrict__ Z6,
               const float* __restrict__ bne_g, const float* __restrict__ bne_b,
               const float* __restrict__ bne_m, const float* __restrict__ bne_v,
               const float* __restrict__ bnr_g, const float* __restrict__ bnr_b,
               const float* __restrict__ bnr_m, const float* __restrict__ bnr_v,
               float* __restrict__ out) {
    extern __shared__ float smem[];
    float* vsh = smem;                    // 7 x 16 BN'd mode vectors
    float* Mb  = smem + 7 * 16;           // 7 ring-core matrices M0..M6
    float* L1b = Mb + 7 * BUFSZ;          // level-1 products P01, P23, P45
    float* red = L1b + 3 * BUFSZ;         // reduction scratch

    const int b    = blockIdx.x;
    const int tid  = threadIdx.x;
    const int wave = tid >> 5;
    const int lane = tid & 31;

    const float* Zs[7] = {Z0, Z1, Z2, Z3, Z4, Z5, Z6};

    // ---- Stage 1: gather + BatchNorm(eval); thread t -> (vector j, dim d) ----
    if (tid < 7 * DE) {
        const int j = tid / DE;
        const int d = tid - j * DE;
        float x, g, bt, mn, vr;
        if (j == 0) {
            x  = R[r_idx[b] * DE + d];
            g  = bnr_g[d]; bt = bnr_b[d]; mn = bnr_m[d]; vr = bnr_v[d];
        } else {
            const int* ep = (j == 1) ? e1_idx : (j == 2) ? e2_idx :
                            (j == 3) ? e3_idx : (j == 4) ? e4_idx :
                            (j == 5) ? e5_idx : e6_idx;
            x  = E[ep[b] * DE + d];
            g  = bne_g[d]; bt = bne_b[d]; mn = bne_m[d]; vr = bne_v[d];
        }
        vsh[j * 16 + d] = g * (x - mn) * rsqrtf(vr + EPS_BN) + bt;
    }
    __syncthreads();

    // ---- Stage 2: build all seven M_j[a,c] = sum_i v_j[i] Z_j[a,i,c] ----
    for (int t = tid; t < 7 * PADM * PADM; t += NTHREADS) {
        const int j  = t / (PADM * PADM);
        const int rr = t - j * (PADM * PADM);
        const int a  = rr / PADM;
        const int c  = rr - a * PADM;
        float acc = 0.f;
        if (a < RANK && c < RANK) {
            const float* zp = Zs[j] + (a * DE) * RANK + c;
            const float* vv = vsh + j * 16;
#pragma unroll
            for (int i = 0; i < DE; ++i)
                acc = fmaf(vv[i], zp[i * RANK], acc);
        }
        Mb[j * BUFSZ + a * STRIDE + c] = acc;
    }
    __syncthreads();

    // ---- Stage 3, level 1: P01=M0@M1, P23=M2@M3, P45=M4@M5 ----
    // 3 matmuls x 9 tiles = 27 independent tiles strided over 4 waves.
    for (int tt = wave; tt < 3 * NT * NT; tt += (NTHREADS / 32)) {
        const int m  = tt / (NT * NT);
        const int tl = tt - m * (NT * NT);
        wmma_tile(L1b + m * BUFSZ,
                  Mb + (2 * m) * BUFSZ, Mb + (2 * m + 1) * BUFSZ,
                  tl / NT, tl - (tl / NT) * NT, lane);
    }
    __syncthreads();

    // ---- Stage 3, level 2: P0123=P01@P23 -> Mb[0]; P456=P45@M6 -> Mb[1] ----
    for (int tt = wave; tt < 2 * NT * NT; tt += (NTHREADS / 32)) {
        const int m  = tt / (NT * NT);
        const int tl = tt - m * (NT * NT);
        const float* A  = (m == 0) ? L1b            : L1b + 2 * BUFSZ;
        const float* Bm = (m == 0) ? L1b + BUFSZ    : Mb + 6 * BUFSZ;
        float*       D  = (m == 0) ? Mb             : Mb + BUFSZ;
        wmma_tile(D, A, Bm, tl / NT, tl - (tl / NT) * NT, lane);
    }
    __syncthreads();

    // ---- Stage 4: pred[b] = trace(P0123 @ P456) = sum P0123[a,c]*P456[c,a] ----
    float acc = 0.f;
    for (int t = tid; t < PADM * PADM; t += NTHREADS) {
        const int a = t / PADM;
        const int c = t - a * PADM;
        if (a < RANK && c < RANK)
            acc = fmaf(Mb[a * STRIDE + c], Mb[BUFSZ + c * STRIDE + a], acc);
    }
    red[tid] = acc;
    __syncthreads();
    if (tid == 0) {
        float s = 0.f;
#pragma unroll
        for (int i = 0; i < NTHREADS; ++i) s += red[i];
        out[b] = s;
    }
}

extern "C" void kernel_launch(void* const* d_in, const int* in_sizes, int n_in,
                              void* d_out, int out_size, void* d_ws, size_t ws_size,
                              hipStream_t stream) {
    const int*   r_idx  = (const int*)d_in[0];
    const int*   e1_idx = (const int*)d_in[1];
    const int*   e2_idx = (const int*)d_in[2];
    const int*   e3_idx = (const int*)d_in[3];
    const int*   e4_idx = (const int*)d_in[4];
    const int*   e5_idx = (const int*)d_in[5];
    const int*   e6_idx = (const int*)d_in[6];
    const float* E      = (const float*)d_in[7];
    const float* R      = (const float*)d_in[8];
    const float* Z0     = (const float*)d_in[9];
    const float* Z1     = (const float*)d_in[10];
    const float* Z2     = (const float*)d_in[11];
    const float* Z3     = (const float*)d_in[12];
    const float* Z4     = (const float*)d_in[13];
    const float* Z5     = (const float*)d_in[14];
    const float* Z6     = (const float*)d_in[15];
    const float* bne_g  = (const float*)d_in[16];
    const float* bne_b  = (const float*)d_in[17];
    const float* bne_m  = (const float*)d_in[18];
    const float* bne_v  = (const float*)d_in[19];
    const float* bnr_g  = (const float*)d_in[20];
    const float* bnr_b  = (const float*)d_in[21];
    const float* bnr_m  = (const float*)d_in[22];
    const float* bnr_v  = (const float*)d_in[23];

    // Allow ~98 KB dynamic LDS (WGP has 320 KB on gfx1250).
    (void)hipFuncSetAttribute((const void*)getd_tr_kernel,
                              hipFuncAttributeMaxDynamicSharedMemorySize,
                              SMEM_BYTES);

    const int B = in_sizes[0];   // 64 batches -> 64 workgroups x 4 wave32
    getd_tr_kernel<<<B, NTHREADS, SMEM_BYTES, stream>>>(
        r_idx, e1_idx, e2_idx, e3_idx, e4_idx, e5_idx, e6_idx,
        E, R, Z0, Z1, Z2, Z3, Z4, Z5, Z6,
        bne_g, bne_b, bne_m, bne_v,
        bnr_g, bnr_b, bnr_m, bnr_v,
        (float*)d_out);
}